// AttentionBlock_10041633538830
// MI455X (gfx1250) — compile-verified
//
#include <hip/hip_runtime.h>

// ---------------------------------------------------------------------------
// AttentionBlock for MI455X (gfx1250, wave32) using native f32 WMMA
//   V_WMMA_F32_16X16X4_F32 : D[16x16] += A[16x4] * B[4x16], all f32.
// Fragment layouts (CDNA5 ISA 7.12.2, 32-bit):
//   A 16x4 : lane L holds M = L%16 ; VGPR j holds K = 2*(L>>4) + j
//   B 4x16 : lane L holds N = L%16 ; VGPR j holds K = 2*(L>>4) + j
//   C 16x16: lane L holds N = L%16 ; VGPR v holds M = v + 8*(L>>4)
// ---------------------------------------------------------------------------

typedef float v2f __attribute__((ext_vector_type(2)));
typedef float v8f __attribute__((ext_vector_type(8)));
typedef int   i32x4 __attribute__((ext_vector_type(4)));

#define BB 16
#define CC 512
#define NN 1024   // h*w
#define NH 4
#define DH 128
#define NG 8

// Async global->LDS (ASYNCcnt-tracked) if this toolchain exposes it.
// Probe round 2 leaked the signature: (int4 AS1*, int4 AS3*, imm, imm).
#if defined(__AMDGCN__) && \
    __has_builtin(__builtin_amdgcn_global_load_async_to_lds_b128) && \
    __has_builtin(__builtin_amdgcn_s_wait_asynccnt)
#define USE_ASYNC_LDS 1
typedef __attribute__((address_space(1))) i32x4 gbl_i4;
typedef __attribute__((address_space(3))) i32x4 lds_i4;
#else
#define USE_ASYNC_LDS 0
#endif

__device__ __forceinline__ v8f wmma_f32(v2f a, v2f b, v8f c) {
    return __builtin_amdgcn_wmma_f32_16x16x4_f32(
        /*neg_a=*/false, a, /*neg_b=*/false, b,
        /*c_mod=*/(short)0, c, /*reuse_a=*/false, /*reuse_b=*/false);
}

// ---------------------------------------------------------------------------
// Kernel 1: GroupNorm(8, 512).  One block per (b, group); group region is
// contiguous: 64 channels * 1024 = 65536 floats.  Memory-bound.
// ---------------------------------------------------------------------------
__global__ __launch_bounds__(256) void groupnorm_kernel(
    const float* __restrict__ x, const float* __restrict__ gamma,
    const float* __restrict__ beta, float* __restrict__ xn) {
    const int gid = blockIdx.x;                  // b*8 + g
    const int g = gid & (NG - 1);
    const int tid = threadIdx.x;
    const int GSZ = (CC / NG) * NN;              // 65536
    const float4* src = (const float4*)(x + (size_t)gid * GSZ);
    float4* dst = (float4*)(xn + (size_t)gid * GSZ);

    float s = 0.f, ss = 0.f;
    for (int i = tid; i < GSZ / 4; i += 256) {
        float4 v = src[i];
        s  += v.x + v.y + v.z + v.w;
        ss += v.x * v.x + v.y * v.y + v.z * v.z + v.w * v.w;
    }
    __shared__ float rs[256], rss[256];
    rs[tid] = s; rss[tid] = ss;
    __syncthreads();
    for (int o = 128; o > 0; o >>= 1) {
        if (tid < o) { rs[tid] += rs[tid + o]; rss[tid] += rss[tid + o]; }
        __syncthreads();
    }
    const float mu   = rs[0] * (1.0f / GSZ);
    const float var  = rss[0] * (1.0f / GSZ) - mu * mu;   // biased variance
    const float rinv = rsqrtf(var + 1e-5f);
    const int cbase  = g * (CC / NG);

    for (int i = tid; i < GSZ / 4; i += 256) {
        float4 v = src[i];
        const int c = cbase + ((i << 2) >> 10);   // 1024 % 4 == 0 -> same channel
        const float gm = gamma[c] * rinv;
        const float bt = beta[c] - mu * gm;
        float4 o4;
        o4.x = v.x * gm + bt; o4.y = v.y * gm + bt;
        o4.z = v.z * gm + bt; o4.w = v.w * gm + bt;
        dst[i] = o4;
    }
}

// ---------------------------------------------------------------------------
// Kernels 2 & 4: batched GEMM  out[b][m][n] = sum_k W[m][k] * Bm[b][k][n]
//                               (+ bias[m]) (+ res[b][m][n])
// N = 1024, K = 512 fixed.  Each wave computes a 64(M) x 64(N) tile as a
// 4x4 grid of 16x16 WMMA accumulators (128 acc VGPRs); 8 waves per block.
// 16 WMMA per k-step vs 16 dwords/lane loaded -> ~16 FLOP/B from L2
// (W is 3MB, one batch of B is 2MB: both L2-resident at 192MB).
// ---------------------------------------------------------------------------
__global__ __launch_bounds__(256, 1) void gemm_kernel(
    const float* __restrict__ W,    // [M][K]
    const float* __restrict__ Bm,   // [batch][K][N]
    const float* __restrict__ bias, // [M]
    const float* __restrict__ res,  // [batch][M][N] or nullptr
    float* __restrict__ out,        // [batch][M][N]
    int M) {
    const int K = CC, N = NN;
    const int lane = threadIdx.x & 31;
    const int l15  = lane & 15;
    const int lh   = lane >> 4;
    const int wid  = blockIdx.x * 8 + (threadIdx.x >> 5);

    const int ntiles = N / 64;
    const int per_b  = (M / 64) * ntiles;
    const int b  = wid / per_b;
    const int r  = wid % per_b;
    const int m0 = (r / ntiles) * 64;
    const int n0 = (r % ntiles) * 64;

    const float* Bp = Bm + (size_t)b * K * N;
    v8f acc[4][4] = {};

    for (int k = 0; k < K; k += 4) {
        const int kr = k + 2 * lh;               // even -> 8B-aligned v2f load
        v2f a[4], bf[4];
#pragma unroll
        for (int mt = 0; mt < 4; ++mt)
            a[mt] = *(const v2f*)(W + (size_t)(m0 + mt * 16 + l15) * K + kr);
#pragma unroll
        for (int nt = 0; nt < 4; ++nt) {
            const float* bp = Bp + (size_t)kr * N + (n0 + nt * 16 + l15);
            bf[nt].x = bp[0];
            bf[nt].y = bp[N];
        }
        if (k + 32 < K)                           // global_prefetch_b8 hint
            __builtin_prefetch(Bp + (size_t)(kr + 32) * N + n0 + l15, 0, 1);
#pragma unroll
        for (int mt = 0; mt < 4; ++mt)
#pragma unroll
            for (int nt = 0; nt < 4; ++nt)
                acc[mt][nt] = wmma_f32(a[mt], bf[nt], acc[mt][nt]);
    }

    const size_t ob = (size_t)b * M * N;
#pragma unroll
    for (int mt = 0; mt < 4; ++mt)
#pragma unroll
        for (int nt = 0; nt < 4; ++nt)
#pragma unroll
            for (int v = 0; v < 8; ++v) {
                const int row = m0 + mt * 16 + v + 8 * lh;
                const int col = n0 + nt * 16 + l15;
                float val = acc[mt][nt][v] + bias[row];
                if (res) val += res[ob + (size_t)row * N + col];
                out[ob + (size_t)row * N + col] = val;
            }
}

// ---------------------------------------------------------------------------
// Kernel 3: flash-style attention.  One block per (b, head, 64-query tile),
// 4 waves; each wave owns a 16-query strip.  Q strip (16x128) lives entirely
// in registers as 32 WMMA A-fragments (reused for all 32 key tiles).
// K/V tiles staged in LDS in d-major layout [128][36]:
//   - global 16B chunks == LDS 16B chunks -> async b128 copies (ASYNCcnt)
//   - S-phase reads lane-contiguous; PV-phase reads stride-36 (36i mod 64
//     distinct for i<16 -> bank-conflict-free).
// Online softmax per row; O (16x128) in 8 WMMA C-tiles.
// ---------------------------------------------------------------------------
__global__ __launch_bounds__(128, 1) void attn_kernel(
    const float* __restrict__ qkv,    // [b][3*C][N]
    float* __restrict__ aout) {       // [b][C][N]  (c = head*DH + d)
    const int bid  = blockIdx.x;
    const int qt   = bid & 15;
    const int head = (bid >> 4) & 3;
    const int b    = bid >> 6;
    const int tid  = threadIdx.x;
    const int wave = tid >> 5, lane = tid & 31;
    const int l15 = lane & 15, lh = lane >> 4;
    const float scale = 0.08838834764831845f;    // 1/sqrt(128)

    const float* Qp = qkv + ((size_t)b * 3 * CC + head * DH) * NN;
    const float* Kp = qkv + ((size_t)b * 3 * CC + CC + head * DH) * NN;
    const float* Vp = qkv + ((size_t)b * 3 * CC + 2 * CC + head * DH) * NN;

    const int q0 = qt * 64 + wave * 16;          // first query of this strip

    // Q fragments: qf[s][j] = Q[m = q0 + L%16][d = 4s + 2*(L>>4) + j]
    v2f qf[32];
#pragma unroll
    for (int s = 0; s < 32; ++s) {
        const int d = 4 * s + 2 * lh;
        const int m = q0 + l15;
        qf[s].x = Qp[(size_t)d * NN + m];
        qf[s].y = Qp[(size_t)(d + 1) * NN + m];
    }

    __shared__ __align__(16) float Ks[DH][36];   // [d][key], 4-float row pad
    __shared__ __align__(16) float Vs[DH][36];
    __shared__ float Sc[4][16][33];              // per-wave scores/probs
    __shared__ float rowf[4][16];                // per-row broadcast factors

    v8f o[8] = {};                               // O strip: 16 x 128 (8 C-tiles)
    float mrow = -3.0e38f, lrow = 0.f;           // row = l15 (lanes 16-31 dup)

    for (int kt = 0; kt < 32; ++kt) {
        __syncthreads();
#if USE_ASYNC_LDS
        // 32 keys x 128 d = 1024 float4 chunks per matrix; 8 per thread.
        for (int i = tid; i < 1024; i += 128) {
            const int d  = i >> 3;               // 8 float4 per d-row
            const int m4 = (i & 7) << 2;         // key within tile (x4)
            __builtin_amdgcn_global_load_async_to_lds_b128(
                (gbl_i4*)(Kp + (size_t)d * NN + kt * 32 + m4),
                (lds_i4*)&Ks[d][m4], 0, 0);
            __builtin_amdgcn_global_load_async_to_lds_b128(
                (gbl_i4*)(Vp + (size_t)d * NN + kt * 32 + m4),
                (lds_i4*)&Vs[d][m4], 0, 0);
        }
        __builtin_amdgcn_s_wait_asynccnt(0);
#else
        for (int i = tid; i < 32 * DH; i += 128) {
            const int m = i & 31, d = i >> 5;    // m fastest: coalesced global
            Ks[d][m] = Kp[(size_t)d * NN + kt * 32 + m];
            Vs[d][m] = Vp[(size_t)d * NN + kt * 32 + m];
        }
#endif
        __syncthreads();

        // S = Q(16x128) * K^T(128x32) : 32 k-steps x 2 key sub-tiles
        v8f s2[2] = {};
#pragma unroll
        for (int s = 0; s < 32; ++s) {
            const int kb = 4 * s + 2 * lh;
#pragma unroll
            for (int t = 0; t < 2; ++t) {
                v2f bf;                          // B[k=d][n=key] = K[key][d]
                bf.x = Ks[kb][t * 16 + l15];
                bf.y = Ks[kb + 1][t * 16 + l15];
                s2[t] = wmma_f32(qf[s], bf, s2[t]);
            }
        }
        // spill S (scaled) to wave-private LDS for row-wise softmax
#pragma unroll
        for (int t = 0; t < 2; ++t)
#pragma unroll
            for (int v = 0; v < 8; ++v)
                Sc[wave][v + 8 * lh][t * 16 + l15] = s2[t][v] * scale;

        // online softmax update; lane L processes row L%16 (lanes 16-31
        // duplicate -> identical values written, EXEC stays all-ones)
        float rmax = -3.0e38f;
        for (int j = 0; j < 32; ++j) rmax = fmaxf(rmax, Sc[wave][l15][j]);
        const float mnew  = fmaxf(mrow, rmax);
        const float alpha = __expf(mrow - mnew);
        float ps = 0.f;
        for (int j = 0; j < 32; ++j) {
            const float p = __expf(Sc[wave][l15][j] - mnew);
            Sc[wave][l15][j] = p;
            ps += p;
        }
        lrow = alpha * lrow + ps;
        mrow = mnew;
        rowf[wave][l15] = alpha;

        // rescale O accumulators: lane holds rows v + 8*(L>>4)
        float fv[8];
#pragma unroll
        for (int v = 0; v < 8; ++v) fv[v] = rowf[wave][v + 8 * lh];
#pragma unroll
        for (int t = 0; t < 8; ++t)
#pragma unroll
            for (int v = 0; v < 8; ++v) o[t][v] *= fv[v];

        // O += P(16x32) * V(32x128) : 8 k-steps x 8 d sub-tiles
#pragma unroll
        for (int s = 0; s < 8; ++s) {
            const int kb = 4 * s + 2 * lh;
            v2f pa;
            pa.x = Sc[wave][l15][kb];
            pa.y = Sc[wave][l15][kb + 1];
#pragma unroll
            for (int t = 0; t < 8; ++t) {
                v2f bf;                          // B[k=key][n=d] = V[key][d]
                bf.x = Vs[t * 16 + l15][kb];
                bf.y = Vs[t * 16 + l15][kb + 1];
                o[t] = wmma_f32(pa, bf, o[t]);
            }
        }
    }

    // final 1/l normalization and store: aout[b][head*128 + d][n]
    rowf[wave][l15] = 1.0f / lrow;
    float fv[8];
#pragma unroll
    for (int v = 0; v < 8; ++v) fv[v] = rowf[wave][v + 8 * lh];
#pragma unroll
    for (int t = 0; t < 8; ++t)
#pragma unroll
        for (int v = 0; v < 8; ++v) {
            const int d = t * 16 + l15;
            const int n = q0 + v + 8 * lh;
            aout[((size_t)b * CC + head * DH + d) * NN + n] = o[t][v] * fv[v];
        }
}

// ---------------------------------------------------------------------------
// Launch: GroupNorm -> QKV GEMM -> attention -> proj GEMM (+bias +residual)
// Workspace: xn (32MB) | qkv (96MB); attention output reuses the xn region.
// ---------------------------------------------------------------------------
extern "C" void kernel_launch(void* const* d_in, const int* in_sizes, int n_in,
                              void* d_out, int out_size, void* d_ws, size_t ws_size,
                              hipStream_t stream) {
    const float* x      = (const float*)d_in[0];
    const float* gamma  = (const float*)d_in[1];
    const float* beta   = (const float*)d_in[2];
    const float* qkv_w  = (const float*)d_in[3];
    const float* qkv_b  = (const float*)d_in[4];
    const float* proj_w = (const float*)d_in[5];
    const float* proj_b = (const float*)d_in[6];
    float* out = (float*)d_out;

    float* xn     = (float*)d_ws;                    // [16][512][1024]
    float* qkvbuf = xn + (size_t)BB * CC * NN;       // [16][1536][1024]
    float* aout   = xn;                              // reuse after QKV GEMM

    groupnorm_kernel<<<BB * NG, 256, 0, stream>>>(x, gamma, beta, xn);

    // QKV: M=1536 -> 16 * 24 * 16 = 6144 wave-tiles / 8 waves = 768 blocks
    gemm_kernel<<<(BB * (3 * CC / 64) * (NN / 64)) / 8, 256, 0, stream>>>(
        qkv_w, xn, qkv_b, nullptr, qkvbuf, 3 * CC);

    // attention: 16 b * 4 heads * 16 query-tiles = 1024 blocks of 4 waves
    attn_kernel<<<BB * NH * 16, 128, 0, stream>>>(qkvbuf, aout);

    // proj + bias + residual: M=512 -> 16 * 8 * 16 = 2048 / 8 = 256 blocks
    gemm_kernel<<<(BB * (CC / 64) * (NN / 64)) / 8, 256, 0, stream>>>(
        proj_w, aout, proj_b, x, out, CC);
}